// ARIGUserEncoder_39926015983975
// MI455X (gfx1250) — compile-verified
//
#include <hip/hip_runtime.h>
#include <hip/hip_bf16.h>
#include <math.h>

#define T_LEN  200
#define D_DIM  128
#define KSHORT 5

#ifndef __has_builtin
#define __has_builtin(x) 0
#endif
#if __has_builtin(__builtin_amdgcn_global_load_async_to_lds_b128)
#define HAVE_ASYNC_LDS 1
#else
#define HAVE_ASYNC_LDS 0
#endif

typedef __attribute__((ext_vector_type(2))) float v2f;
typedef __attribute__((ext_vector_type(8))) float v8f;

// ---------------------------------------------------------------------------
// Kernel 0: M = Wq^T @ Wk  (128x128x128) via V_WMMA_F32_16X16X4_F32.
// One wave per 16x16 output tile; 32 WMMA steps over K (K=4 each).
// ---------------------------------------------------------------------------
__global__ __launch_bounds__(32) void wqk_gemm_kernel(const float* __restrict__ Wq,
                                                      const float* __restrict__ Wk,
                                                      float* __restrict__ Mout) {
  const int lane  = threadIdx.x;
  const int n     = lane & 15;
  const int khalf = lane >> 4;       // 0 or 1
  const int d0 = blockIdx.x * 16;    // output col tile
  const int f0 = blockIdx.y * 16;    // output row tile
  v8f c = {};
  for (int e0 = 0; e0 < D_DIM; e0 += 4) {
    v2f a, b;
    // A[m][k] = (Wq^T)[f0+m][e0+k] = Wq[e0+k][f0+m]
    a.x = Wq[(e0 + 2 * khalf + 0) * D_DIM + f0 + n];
    a.y = Wq[(e0 + 2 * khalf + 1) * D_DIM + f0 + n];
    // B[k][nn] = Wk[e0+k][d0+nn]
    b.x = Wk[(e0 + 2 * khalf + 0) * D_DIM + d0 + n];
    b.y = Wk[(e0 + 2 * khalf + 1) * D_DIM + d0 + n];
    c = __builtin_amdgcn_wmma_f32_16x16x4_f32(false, a, false, b,
                                              (short)0, c, false, false);
  }
#pragma unroll
  for (int i = 0; i < 8; ++i) {
    Mout[(f0 + i + 8 * khalf) * D_DIM + d0 + n] = c[i];
  }
}

// ---------------------------------------------------------------------------
// Block-wide reductions (256 threads).
// ---------------------------------------------------------------------------
__device__ __forceinline__ float block_sum(float v, float* red, int tid) {
  red[tid] = v; __syncthreads();
#pragma unroll
  for (int s = 128; s > 0; s >>= 1) {
    if (tid < s) red[tid] += red[tid + s];
    __syncthreads();
  }
  float r = red[0]; __syncthreads();
  return r;
}
__device__ __forceinline__ float block_max(float v, float* red, int tid) {
  red[tid] = v; __syncthreads();
#pragma unroll
  for (int s = 128; s > 0; s >>= 1) {
    if (tid < s) red[tid] = fmaxf(red[tid], red[tid + s]);
    __syncthreads();
  }
  float r = red[0]; __syncthreads();
  return r;
}

// ---------------------------------------------------------------------------
// Kernel A: one workgroup (256 threads = 8 waves) per batch row b.
// Single HBM pass: x_b staged into LDS via GLOBAL_LOAD_ASYNC_TO_LDS_B128.
// LDS budget ~106.6KB -> 3 workgroups resident per 320KB WGP.
// ---------------------------------------------------------------------------
__global__ __launch_bounds__(256) void user_encoder_kernel(
    const float* __restrict__ hist,          // (B,T,D)
    const unsigned char* __restrict__ mask,  // (B,T) bool
    const float* __restrict__ age,           // (B,T)
    const float* __restrict__ pop,           // (B,T)
    const float* __restrict__ Mqk,           // (D,D) = Wq^T @ Wk
    const float* __restrict__ Wv,            // (D,D)
    const float* __restrict__ gate_w,        // (2,)
    const float* __restrict__ gate_b,        // (1,)
    const float* __restrict__ ln_g,          // (D,)
    const float* __restrict__ ln_b,          // (D,)
    const float* __restrict__ dalpha,        // (1,)
    float* __restrict__ out)                 // (B,D)
{
  __shared__ float xs[T_LEN * D_DIM];  // 102400 B
  __shared__ float maskf[T_LEN];
  __shared__ float sc[T_LEN];
  __shared__ float red[256];
  __shared__ float meanv[D_DIM];
  __shared__ float qkv[D_DIM];
  __shared__ float wsv[D_DIM];
  __shared__ float scal[4];

  const int b   = blockIdx.x;
  const int tid = threadIdx.x;

  // ---- Phase 1: stage x_b into LDS (coalesced b128), plus mask vector.
  {
    const float4* xg = reinterpret_cast<const float4*>(hist + (size_t)b * T_LEN * D_DIM);
    float4* xl = reinterpret_cast<float4*>(xs);
    const int nvec = T_LEN * D_DIM / 4;  // 6400
#if HAVE_ASYNC_LDS
    typedef int v4i __attribute__((vector_size(16)));
    typedef __attribute__((address_space(1))) v4i* gptr_t;
    typedef __attribute__((address_space(3))) v4i* lptr_t;
    for (int i = tid; i < nvec; i += 256) {
      __builtin_amdgcn_global_load_async_to_lds_b128(
          (gptr_t)(xg + i), (lptr_t)(xl + i), 0, 0);
    }
#else
    for (int i = tid; i < nvec; i += 256) xl[i] = xg[i];
#endif
  }
  if (tid < T_LEN) {
    maskf[tid] = mask[(size_t)b * T_LEN + tid] ? 1.0f : 0.0f;
  }
#if HAVE_ASYNC_LDS
#if __has_builtin(__builtin_amdgcn_s_wait_asynccnt)
  __builtin_amdgcn_s_wait_asynccnt(0);
#else
  asm volatile("s_wait_asynccnt 0x0" ::: "memory");
#endif
#endif
  __syncthreads();

  // ---- Phase 2: mask count.
  float cntf = block_sum((tid < T_LEN) ? maskf[tid] : 0.0f, red, tid);
  int cnti = (int)(cntf + 0.5f);
  if (cnti < 1) cnti = 1;

  // ---- Phase 3: masked mean over T (lane->bank identity, conflict-free),
  //               and gate scalar from the last-K window (<=5 global reads).
  if (tid < D_DIM) {
    float s = 0.f;
    for (int t = 0; t < T_LEN; ++t) s += maskf[t] * xs[t * D_DIM + tid];
    meanv[tid] = s / (cntf + 1e-6f);
  }
  if (tid == 0) {
    int t0 = cnti - KSHORT; if (t0 < 0) t0 = 0;
    float dn = (float)(cnti - t0); if (dn < 1.f) dn = 1.f;
    float sp = 0.f, sr = 0.f;
    for (int t = t0; t < cnti; ++t) {
      sp += pop[(size_t)b * T_LEN + t];
      sr += age[(size_t)b * T_LEN + t];
    }
    float z = (sp / dn) * gate_w[0] + (sr / dn) * gate_w[1] + gate_b[0];
    scal[0] = 1.0f / (1.0f + expf(-z));   // g
    scal[1] = dn;
    scal[2] = (float)t0;
  }
  __syncthreads();

  // ---- Phase 4: qk = mean @ M  (M is 64KB, L2-hot; coalesced per column).
  if (tid < D_DIM) {
    float s = 0.f;
    for (int f = 0; f < D_DIM; ++f) s += meanv[f] * Mqk[f * D_DIM + tid];
    qkv[tid] = s;
  }
  __syncthreads();

  // ---- Phase 5: scores_t = scale * (x_t . qk) + log(exp(-alpha*age)+1e-12),
  //               masked. Skewed index -> conflict-free LDS row walks.
  const float alpha = log1pf(expf(dalpha[0])) + 1e-6f;
  const float scale = 0.08838834764831845f;  // 1/sqrt(128)
  if (tid < T_LEN) {
    float s = 0.f;
    const int base = tid * D_DIM;
    for (int i = 0; i < D_DIM; ++i) {
      const int d = (tid + i) & (D_DIM - 1);
      s += xs[base + d] * qkv[d];
    }
    const float a_t = age[(size_t)b * T_LEN + tid];
    const float dec = expf(-alpha * a_t);
    float sco = s * scale + logf(dec + 1e-12f);
    sc[tid] = (maskf[tid] > 0.f) ? sco : -3.0e38f;
  }
  __syncthreads();

  // ---- Phase 6: softmax over T.
  float mx = block_max((tid < T_LEN) ? sc[tid] : -3.0e38f, red, tid);
  float ev = (tid < T_LEN) ? expf(sc[tid] - mx) : 0.0f;
  float ssum = block_sum(ev, red, tid);
  if (tid < T_LEN) sc[tid] = ev / ssum;
  __syncthreads();

  // ---- Phase 7: wsum_d = sum_t attn_t * x_td.
  if (tid < D_DIM) {
    float s = 0.f;
    for (int t = 0; t < T_LEN; ++t) s += sc[t] * xs[t * D_DIM + tid];
    wsv[tid] = s;
  }
  __syncthreads();

  // ---- Phase 8: long = wsum @ Wv^T, short = last-K mean, gate mix, LN.
  float u = 0.f;
  const int t0 = (int)scal[2];
  if (tid < D_DIM) {
    float lt = 0.f;
    const float* wrow = Wv + (size_t)tid * D_DIM;
    for (int d = 0; d < D_DIM; ++d) lt += wsv[d] * wrow[d];
    float st = 0.f;
    for (int t = t0; t < cnti; ++t) st += xs[t * D_DIM + tid];
    st /= scal[1];
    const float g = scal[0];
    u = g * st + (1.f - g) * lt;
  }
  float mu = block_sum((tid < D_DIM) ? u : 0.f, red, tid) * (1.0f / D_DIM);
  float va = block_sum((tid < D_DIM) ? (u - mu) * (u - mu) : 0.f, red, tid) * (1.0f / D_DIM);
  if (tid < D_DIM) {
    out[(size_t)b * D_DIM + tid] =
        (u - mu) * rsqrtf(va + 1e-5f) * ln_g[tid] + ln_b[tid];
  }
}

// ---------------------------------------------------------------------------
extern "C" void kernel_launch(void* const* d_in, const int* in_sizes, int n_in,
                              void* d_out, int out_size, void* d_ws, size_t ws_size,
                              hipStream_t stream) {
  const float*         hist = (const float*)d_in[0];
  const unsigned char* mask = (const unsigned char*)d_in[1];
  const float*         age  = (const float*)d_in[2];
  const float*         pop  = (const float*)d_in[3];
  const float*         Wq   = (const float*)d_in[4];
  const float*         Wk   = (const float*)d_in[5];
  const float*         Wv   = (const float*)d_in[6];
  const float*         gw   = (const float*)d_in[7];
  const float*         gb   = (const float*)d_in[8];
  const float*         lg   = (const float*)d_in[9];
  const float*         lb   = (const float*)d_in[10];
  const float*         da   = (const float*)d_in[11];
  float* out = (float*)d_out;
  float* Mws = (float*)d_ws;  // 128*128 f32 = 64KB scratch for Wq^T@Wk

  // Tiny exact-f32 WMMA GEMM: M = Wq^T @ Wk (8x8 tiles of 16x16, 1 wave each).
  dim3 g0(D_DIM / 16, D_DIM / 16);
  wqk_gemm_kernel<<<g0, 32, 0, stream>>>(Wq, Wk, Mws);

  const int B = in_sizes[0] / (T_LEN * D_DIM);  // 4096
  user_encoder_kernel<<<B, 256, 0, stream>>>(hist, mask, age, pop, Mws, Wv,
                                             gw, gb, lg, lb, da, out);
}